// calc_padding_42786464202786
// MI455X (gfx1250) — compile-verified
//
#include <hip/hip_runtime.h>

typedef float v2f __attribute__((ext_vector_type(2)));
typedef float v8f __attribute__((ext_vector_type(8)));

#define BATCH 16
#define CHN   256
#define HIN   128
#define WIN   128
#define HP    132
#define WP    132
#define BC    (BATCH * CHN)   // 4096 (b,c) images
#define IMG   (HP * WP)       // 17424 floats per padded image
#define TILES 9               // ceil(130 outputs / 16 per WMMA tile)

struct EdgeJob {
  const float *W1, *b1, *W2, *b2, *W3, *b3;
  int in0, in1, out, st;      // float offsets within one HPxWP image; st = strip stride
};

// ---------------------------------------------------------------------------
// Pad/copy: out[b,c,h,w] = interior ? x[b,c,h-2,w-2] : 0.  Memory-bound
// (~554 MB total traffic -> ~24us floor at 23.3 TB/s).  float4 (b128) stores,
// rows are 132 floats = 528 B so every float4 store is 16B-aligned.
// ---------------------------------------------------------------------------
__global__ __launch_bounds__(256) void pad_kernel(const float* __restrict__ x,
                                                  float* __restrict__ out,
                                                  long long n4) {
  long long idx = (long long)blockIdx.x * 256 + threadIdx.x;
  if (idx >= n4) return;
  int w4       = (int)(idx % 33);          // 33 float4 per 132-float row
  long long t  = idx / 33;
  int h        = (int)(t % HP);
  long long bc = t / HP;
  float4 v; v.x = v.y = v.z = v.w = 0.f;
  if (h >= 2 && h < 2 + HIN) {
    const float* xr = x + (bc * HIN + (h - 2)) * WIN;
    int c0 = w4 * 4;
    float tmp[4];
#pragma unroll
    for (int k = 0; k < 4; ++k) {
      int c = c0 + k;
      tmp[k] = (c >= 2 && c < 2 + WIN) ? xr[c - 2] : 0.f;
    }
    v.x = tmp[0]; v.y = tmp[1]; v.z = tmp[2]; v.w = tmp[3];
  }
  reinterpret_cast<float4*>(out)[idx] = v;
}

// ---------------------------------------------------------------------------
// Edge MLP: one wave handles one 16-position tile of one strip of one (b,c)
// image, for one of two fused edges.  Layers 1 and 2 ride on
// V_WMMA_F32_16X16X4_F32 (exact fp32 math, matches reference numerics):
//   L1: D = W1(16x4|16x4) x F(4x16|4x16) + b1   (K=6 padded to 8, two WMMAs)
//   L2: D = W2 x relu(L1) + b2                  (K=8, two WMMAs; B operand
//       built from the D layout with two lane-shuffles per K-block)
//   L3: per-lane 8-FMA dot with W3 (cheaper than a third WMMA).
// Rows 8..15 of every tile are exactly zero (zero A rows + zero C rows +
// ReLU), so only lanes 0..15 carry outputs.  EXEC is full at every WMMA.
// ---------------------------------------------------------------------------
__global__ __launch_bounds__(256) void edge_mlp(float* __restrict__ xp,
                                                EdgeJob e0, EdgeJob e1) {
  const int lane = threadIdx.x & 31;
  const int wid  = blockIdx.x * (256 / 32) + (threadIdx.x >> 5);
  const int tile = wid % TILES;
  const int rest = wid / TILES;
  const int bc   = rest & (BC - 1);
  const int eg   = rest >> 12;            // wave-uniform edge select
  const EdgeJob e = eg ? e1 : e0;

  const int m  = lane & 15;               // matrix row (A) or column (B/C/D)
  const int hi = lane >> 4;               // lane-half selects K/M block

  // ---- A operands: W1 (8x6 -> 16x8 zero-padded), W2 (8x8) in A layout ----
  float a1a0 = 0.f, a1a1 = 0.f, a1b0 = 0.f, a1b1 = 0.f;
  float a2a0 = 0.f, a2a1 = 0.f, a2b0 = 0.f, a2b1 = 0.f;
  if (m < 8) {
    const float* w1 = e.W1 + m * 6;       // W1[o][i][kh], k = i*3+kh
    a1a0 = hi ? w1[2] : w1[0];
    a1a1 = hi ? w1[3] : w1[1];
    a1b0 = hi ? 0.f   : w1[4];
    a1b1 = hi ? 0.f   : w1[5];
    const float* w2 = e.W2 + m * 8;
    a2a0 = hi ? w2[2] : w2[0];
    a2a1 = hi ? w2[3] : w2[1];
    a2b0 = hi ? w2[6] : w2[4];
    a2b1 = hi ? w2[7] : w2[5];
  }
  float c1v[8], c2v[8], w3v[8];
#pragma unroll
  for (int v = 0; v < 8; ++v) {
    float b1 = e.b1[v], b2 = e.b2[v];
    c1v[v] = hi ? 0.f : b1;               // bias only in rows 0..7
    c2v[v] = hi ? 0.f : b2;
    w3v[v] = e.W3[v];
  }
  const float b3s = e.b3[0];

  // ---- feature gather (B operand of layer 1) ----
  float* img = xp + (long long)bc * IMG;
  const int j  = tile * 16 + m;           // output index along strip, 0..143
  const int jc = j > 129 ? 129 : j;       // clamp so loads stay in-bounds
  const int st = e.st;
  const float* p0 = img + e.in0;
  const float* p1 = img + e.in1;
  // f_k(pos) = strip[k/3][pos + k%3];  B layout: lo lanes K0/K1, hi lanes K2/K3
  float b1a0, b1a1, b1b0, b1b1;
  if (!hi) {
    b1a0 = p0[jc * st];                   // f0
    b1a1 = p0[(jc + 1) * st];             // f1
    b1b0 = p1[(jc + 1) * st];             // f4
    b1b1 = p1[(jc + 2) * st];             // f5
  } else {
    b1a0 = p0[(jc + 2) * st];             // f2
    b1a1 = p1[jc * st];                   // f3
    b1b0 = 0.f;                           // K=6 pad
    b1b1 = 0.f;                           // K=7 pad
  }

  // ---- layer 1: two K=4 fp32 WMMAs + ReLU ----
  v2f A1a = {a1a0, a1a1}, B1a = {b1a0, b1a1};
  v2f A1b = {a1b0, a1b1}, B1b = {b1b0, b1b1};
  v8f acc;
#pragma unroll
  for (int v = 0; v < 8; ++v) acc[v] = c1v[v];
  acc = __builtin_amdgcn_wmma_f32_16x16x4_f32(false, A1a, false, B1a,
                                              (short)0, acc, false, false);
  acc = __builtin_amdgcn_wmma_f32_16x16x4_f32(false, A1b, false, B1b,
                                              (short)0, acc, false, false);
#pragma unroll
  for (int v = 0; v < 8; ++v) acc[v] = fmaxf(acc[v], 0.f);

  // ---- D-layout -> B-layout conversion for layer 2 ----
  // hi lanes need H1 rows 2/3 (resp. 6/7) of column (lane-16) = lo lane regs
  const int src = lane & 15;
  float t2 = __shfl(acc[2], src, 32);
  float t3 = __shfl(acc[3], src, 32);
  float t6 = __shfl(acc[6], src, 32);
  float t7 = __shfl(acc[7], src, 32);

  v2f A2a = {a2a0, a2a1};
  v2f B2a = {hi ? t2 : acc[0], hi ? t3 : acc[1]};
  v2f A2b = {a2b0, a2b1};
  v2f B2b = {hi ? t6 : acc[4], hi ? t7 : acc[5]};
  v8f acc2;
#pragma unroll
  for (int v = 0; v < 8; ++v) acc2[v] = c2v[v];
  acc2 = __builtin_amdgcn_wmma_f32_16x16x4_f32(false, A2a, false, B2a,
                                               (short)0, acc2, false, false);
  acc2 = __builtin_amdgcn_wmma_f32_16x16x4_f32(false, A2b, false, B2b,
                                               (short)0, acc2, false, false);

  // ---- layer 3: per-lane dot (ReLU fused), store only valid lo lanes ----
  float o = b3s;
#pragma unroll
  for (int v = 0; v < 8; ++v) o = fmaf(w3v[v], fmaxf(acc2[v], 0.f), o);
  o = fmaxf(o, 0.f);
  if (!hi && j < 130) img[e.out + j * st] = o;
}

// ---------------------------------------------------------------------------
// Host side
// ---------------------------------------------------------------------------
struct Mlp { const float *W1, *b1, *W2, *b2, *W3, *b3; };

extern "C" void kernel_launch(void* const* d_in, const int* in_sizes, int n_in,
                              void* d_out, int out_size, void* d_ws, size_t ws_size,
                              hipStream_t stream) {
  (void)out_size; (void)d_ws; (void)ws_size;

  // Locate x by its unique element count.
  int xi = -1;
  for (int i = 0; i < n_in; ++i)
    if (in_sizes[i] == BATCH * CHN * HIN * WIN) { xi = i; break; }
  if (xi < 0) xi = 0;
  const float* x = (const float*)d_in[xi];

  // Map the 24 parameter leaves robustly: detect insertion-order
  // (W1,b1,W2,b2,W3,b3 per edge; edges top,bottom,left,right) vs jax
  // sorted-key flatten (W1,W2,W3,b1,b2,b3; edges bottom,left,right,top)
  // by whether the entry after W1(48) is W2(64).  Blob fallback for n_in==2.
  Mlp mlps[4]; // order: top, bottom, left, right
  if (n_in >= 25) {
    int pl[24]; int np = 0;
    for (int i = 0; i < n_in && np < 24; ++i) if (i != xi) pl[np++] = i;
    const bool sorted_keys = (in_sizes[pl[1]] == 64);
    static const int remap_sorted[4] = {1, 2, 3, 0}; // bottom,left,right,top
    for (int g = 0; g < 4; ++g) {
      const int* q = pl + g * 6;
      Mlp m; int dst;
      if (!sorted_keys) {
        m.W1 = (const float*)d_in[q[0]]; m.b1 = (const float*)d_in[q[1]];
        m.W2 = (const float*)d_in[q[2]]; m.b2 = (const float*)d_in[q[3]];
        m.W3 = (const float*)d_in[q[4]]; m.b3 = (const float*)d_in[q[5]];
        dst = g;
      } else {
        m.W1 = (const float*)d_in[q[0]]; m.W2 = (const float*)d_in[q[1]];
        m.W3 = (const float*)d_in[q[2]]; m.b1 = (const float*)d_in[q[3]];
        m.b2 = (const float*)d_in[q[4]]; m.b3 = (const float*)d_in[q[5]];
        dst = remap_sorted[g];
      }
      mlps[dst] = m;
    }
  } else {
    const float* blob = nullptr;
    for (int i = 0; i < n_in; ++i) if (i != xi) { blob = (const float*)d_in[i]; break; }
    static const int remap_sorted[4] = {1, 2, 3, 0};
    long long off = 0;
    for (int g = 0; g < 4; ++g) {
      Mlp m;
      m.W1 = blob + off; off += 48;
      m.W2 = blob + off; off += 64;
      m.W3 = blob + off; off += 8;
      m.b1 = blob + off; off += 8;
      m.b2 = blob + off; off += 8;
      m.b3 = blob + off; off += 1;
      mlps[remap_sorted[g]] = m;
    }
  }

  float* out = (float*)d_out;

  // 1) pad/copy (writes every element of d_out -> deterministic under replay)
  const long long n4 = (long long)BC * HP * (WP / 4);
  const int padBlocks = (int)((n4 + 255) / 256);
  pad_kernel<<<padBlocks, 256, 0, stream>>>(x, out, n4);

  auto mkJob = [](const Mlp& m, int in0, int in1, int o, int st) {
    EdgeJob j;
    j.W1 = m.W1; j.b1 = m.b1; j.W2 = m.W2; j.b2 = m.b2; j.W3 = m.W3; j.b3 = m.b3;
    j.in0 = in0; j.in1 = in1; j.out = o; j.st = st;
    return j;
  };

  // 2) sequential ring growth; stream order reproduces top/bottom -> left/right
  const int egrid = 2 * BC * TILES / 8; // 73728 waves / 8 waves-per-block
  for (int i = 0; i < 2; ++i) {
    EdgeJob jt = mkJob(mlps[0], (2 - i) * WP, (3 - i) * WP, (1 - i) * WP + 1, 1);
    EdgeJob jb = mkJob(mlps[1], (127 + i) * WP, (128 + i) * WP, (130 + i) * WP + 1, 1);
    edge_mlp<<<egrid, 256, 0, stream>>>(out, jt, jb);
    EdgeJob jl = mkJob(mlps[2], 2 - i, 3 - i, WP + (1 - i), WP);
    EdgeJob jr = mkJob(mlps[3], 127 + i, 128 + i, WP + (130 + i), WP);
    edge_mlp<<<egrid, 256, 0, stream>>>(out, jl, jr);
  }
}